// ST_sampling_84842783965715
// MI455X (gfx1250) — compile-verified
//
#include <hip/hip_runtime.h>
#include <math.h>

// ---------------------------------------------------------------------------
// Per (b,c) pair (4096 blocks):
//   Win[300][16]  : 16 gathered windows of x          (staged in LDS)
//   Basis[80][300]: row r = cos(2*pi*(7+bin)*t/300 + phi0)
//                   phi0 = 0 (rows 0..34, cos), -pi/2 (rows 35..69, sin);
//                   rows 70..79 are pad whose results are never read.
//   G = Basis x Win        via v_wmma_f32_16x16x4_f32, K-loop over time
//   p[r][k] = G[r][k]^2 + G[r+35][k]^2 ; out = p / sum_r(p)
// Mean-subtraction touches only DFT bin 0 (outside band 7..41) and the
// forward-norm 1/N cancels in the ratio -> both skipped exactly.
// Roofline: ~156 MB traffic -> ~7 us at 23.3 TB/s; x (147 MB) is L2-resident
// (192 MB L2). Compute ~3 GFLOP of f32 WMMA: far below the memory floor.
// ---------------------------------------------------------------------------

typedef __attribute__((ext_vector_type(2))) float v2f;
typedef __attribute__((ext_vector_type(8))) float v8f;

#define DELTA_T   300
#define KWIN      16
#define NBINS     35      // selected rFFT bins 7..41
#define SEL_LO    7
#define NROWS     70      // 35 cos + 35 sin
#define NTILES    5       // 80 padded rows / 16
#define TPB       (NTILES * 32)   // 160 threads = 5 wave32, one M-tile each
#define WSTRIDE   17      // LDS row stride (floats), odd -> conflict-free

__global__ __launch_bounds__(TPB)
void stft_band_power_kernel(const float* __restrict__ x,
                            const int*   __restrict__ offsets,
                            float*       __restrict__ out,
                            int T)
{
    __shared__ float winT[DELTA_T * WSTRIDE];          // [t][k], padded
    __shared__ float G[NTILES * 16][KWIN + 1];         // GEMM result, padded

    const int bc  = blockIdx.x;                        // (b*C + c)
    const int tid = threadIdx.x;
    const long xbase = (long)bc * (long)T;
    const int* off = offsets + bc * KWIN;

    // ---- Stage 16 windows into LDS, transposed [t][k] -------------------
    // Consecutive tids walk consecutive t within one window -> coalesced
    // global loads; odd LDS stride spreads writes across all banks.
    #pragma unroll
    for (int k = 0; k < KWIN; ++k) {
        const float* src = x + xbase + (long)off[k];
        for (int t = tid; t < DELTA_T; t += TPB) {
            winT[t * WSTRIDE + k] = src[t];
        }
    }
    __syncthreads();

    // ---- Per-wave WMMA: 16-row basis tile x all 16 windows --------------
    const int wave = tid >> 5;           // M-tile index 0..4
    const int lane = tid & 31;
    const int half = lane >> 4;          // f32 A/B layout K-phase split
    const int l16  = lane & 15;
    const int row  = wave * 16 + l16;    // A-matrix row (M = lane%16)

    // Pad rows (>= NROWS) produce garbage that is never read: no masking.
    const bool use_sin = (row >= NBINS) && (row < NROWS);
    const int  bin     = (row < NROWS) ? (use_sin ? row - NBINS : row) : 0;
    const float omega  = 6.283185307179586f * (float)(SEL_LO + bin)
                         / (float)DELTA_T;
    const float phi0   = use_sin ? -1.5707963267948966f : 0.0f;
    const int   phase  = half * 2;       // lanes 0-15: K={0,1}; 16-31: {2,3}

    v8f acc = {};                        // f32 C/D accumulator (8 VGPRs)

    float tf = (float)phase;             // exact fp32 walk: phase, phase+4, ...
    for (int t0 = 0; t0 < DELTA_T; t0 += 4, tf += 4.0f) {
        // A operand: basis(row, t) = cos(omega*t + phi0) at t, t+1
        const float arg0 = fmaf(omega, tf, phi0);
        v2f a;
        a.x = __cosf(arg0);
        a.y = __cosf(arg0 + omega);
        // B operand: Win[t0+phase][n], Win[t0+phase+1][n], n = lane%16
        v2f b;
        b.x = winT[(t0 + phase)     * WSTRIDE + l16];
        b.y = winT[(t0 + phase + 1) * WSTRIDE + l16];
        // D = A(16x4) * B(4x16) + C   -> v_wmma_f32_16x16x4_f32
        acc = __builtin_amdgcn_wmma_f32_16x16x4_f32(
                  false, a, false, b, (short)0, acc, false, false);
    }

    // ---- Spill D tile to LDS: VGPR j -> M = 16*wave + j + 8*half --------
    #pragma unroll
    for (int j = 0; j < 8; ++j) {
        G[wave * 16 + j + half * 8][l16] = acc[j];
    }
    __syncthreads();

    // ---- Power + band normalization; thread k owns window k -------------
    if (tid < KWIN) {
        float p[NBINS];
        float sum = 0.0f;
        #pragma unroll
        for (int r = 0; r < NBINS; ++r) {
            float cr = G[r][tid];
            float si = G[r + NBINS][tid];
            float v  = cr * cr + si * si;
            p[r] = v;
            sum += v;
        }
        float inv = 1.0f / sum;
        float* o = out + ((long)bc * KWIN + tid) * NBINS;
        #pragma unroll
        for (int r = 0; r < NBINS; ++r) o[r] = p[r] * inv;
    }
}

extern "C" void kernel_launch(void* const* d_in, const int* in_sizes, int n_in,
                              void* d_out, int out_size, void* d_ws, size_t ws_size,
                              hipStream_t stream)
{
    const float* x       = (const float*)d_in[0];   // [B][C][T] f32
    const int*   offsets = (const int*)  d_in[1];   // [B][C][K] i32
    float*       out     = (float*)      d_out;     // [B][C][K][35] f32

    const int BC = in_sizes[1] / KWIN;              // B*C = 4096
    const int T  = (int)(in_sizes[0] / BC);         // 9000

    stft_band_power_kernel<<<BC, TPB, 0, stream>>>(x, offsets, out, T);
}